// Sinkhorn_137438954107
// MI455X (gfx1250) — compile-verified
//
#include <hip/hip_runtime.h>

// Sinkhorn (log-space) for B x 64 x 64 f32, 20 iterations, then exp.
// One matrix per 64 threads; column/row held entirely in VGPRs; ping-pong
// padded LDS buffers for the per-iteration transpose (2 barriers/iter).

#define N 64
#define PAD 68            // row stride in dwords: both transpose directions conflict-free
#define ITERS 20
#define MPB 2             // matrices per block -> 128 threads, ~68 KB LDS (2 buffers)

#define LOG2E 1.44269504088896340736f

#if defined(__AMDGCN__) && __has_builtin(__builtin_amdgcn_global_load_async_to_lds_b128) && \
    __has_builtin(__builtin_amdgcn_s_wait_asynccnt)
#define USE_ASYNC_LDS 1
#else
#define USE_ASYNC_LDS 0
#endif

#if USE_ASYNC_LDS
// Builtin pointer params are v4i* with address spaces (arg0 = global source,
// arg1 = LDS dest). Cast through uintptr_t: AS1 shares the flat
// representation; AS3 is the low-32-bit LDS byte offset.
typedef int v4i_t __attribute__((vector_size(16)));
#define GAS_V4(p) ((__attribute__((address_space(1))) v4i_t*)(uintptr_t)(p))
#define LAS_V4(p) ((__attribute__((address_space(3))) v4i_t*)(uintptr_t)(p))
#endif

// In-register log2-softmax over 64 values (input already scaled by log2(e)):
//   u <- u - max - log2( sum 2^(u - max) )   ( == log2e * log_softmax(x) )
__device__ __forceinline__ void log2softmax64(float (&v)[N]) {
  float mx = v[0];
#pragma unroll
  for (int i = 1; i < N; ++i) mx = fmaxf(mx, v[i]);
  float s = 0.0f;
#pragma unroll
  for (int i = 0; i < N; ++i) s += __builtin_amdgcn_exp2f(v[i] - mx);
  const float lse = mx + __builtin_amdgcn_logf(s);  // v_log_f32 = log2
#pragma unroll
  for (int i = 0; i < N; ++i) v[i] -= lse;
}

__global__ __launch_bounds__(MPB * N) void sinkhorn_kernel(
    const float* __restrict__ x, float* __restrict__ out, int nmat) {
  // [matrix][ping-pong buffer][padded tile]
  __shared__ __align__(16) float lds[MPB][2][N * PAD];

  const int tid = threadIdx.x;
  const int t = tid & (N - 1);          // row/column id within the matrix
  int mat = blockIdx.x * MPB + (tid >> 6);
  if (mat >= nmat) mat = nmat - 1;      // benign duplicate work on ragged tail
  const float* __restrict__ src = x + (size_t)mat * (N * N);
  float* __restrict__ dst = out + (size_t)mat * (N * N);
  float* smA = lds[tid >> 6][0];        // target of column-scatter
  float* smB = lds[tid >> 6][1];        // target of row-scatter

  float v[N];  // thread-resident column (later: row) of the matrix

#if USE_ASYNC_LDS
  // Async DMA the 16 KB tile into padded LDS rows (16B chunks, per-lane addrs),
  // then gather column t into registers (conflict-free: bank = (4r+t) & 63).
#pragma unroll
  for (int i = 0; i < 16; ++i) {
    const int chunk = i * 64 + t;       // 1024 float4 chunks per matrix
    const int r = chunk >> 4;
    const int c = (chunk & 15) << 2;
    __builtin_amdgcn_global_load_async_to_lds_b128(
        GAS_V4(src + r * N + c), LAS_V4(smA + r * PAD + c), 0, 0);
  }
  __builtin_amdgcn_s_wait_asynccnt(0);
  __syncthreads();
#pragma unroll
  for (int r = 0; r < N; ++r) v[r] = smA[r * PAD + t] * LOG2E;
  // No barrier needed: the first writes to smA below touch exactly the
  // addresses this same thread just read (column t).
#else
  // Coalesced: at each r, consecutive t read consecutive global dwords.
#pragma unroll
  for (int r = 0; r < N; ++r) v[r] = src[r * N + t] * LOG2E;
#endif

  for (int it = 0; it < ITERS; ++it) {
    // ---- axis=-2: thread t owns column t; reduce over rows in registers ----
    log2softmax64(v);

    // transpose via buffer A: scatter column (stride PAD), gather row (b128).
    // WAR vs. last iteration's A-reads is ordered by the B-buffer barrier.
#pragma unroll
    for (int r = 0; r < N; ++r) smA[r * PAD + t] = v[r];
    __syncthreads();
#pragma unroll
    for (int c = 0; c < N; c += 4) {
      const float4 f = *reinterpret_cast<const float4*>(smA + t * PAD + c);
      v[c + 0] = f.x; v[c + 1] = f.y; v[c + 2] = f.z; v[c + 3] = f.w;
    }

    // ---- axis=-1: thread t owns row t; reduce over columns in registers ----
    log2softmax64(v);

    if (it != ITERS - 1) {
      // transpose back via buffer B: scatter row (b128), gather column.
#pragma unroll
      for (int c = 0; c < N; c += 4) {
        *reinterpret_cast<float4*>(smB + t * PAD + c) =
            make_float4(v[c], v[c + 1], v[c + 2], v[c + 3]);
      }
      __syncthreads();
#pragma unroll
      for (int r = 0; r < N; ++r) v[r] = smB[r * PAD + t];
      // next write targets A -> no extra barrier needed here (see above)
    } else {
      // Final iteration: v = log2e * log_softmax, so exp(result) == exp2(v).
      // Fuse exp + output transpose into the last transpose-back.
#pragma unroll
      for (int c = 0; c < N; ++c) v[c] = __builtin_amdgcn_exp2f(v[c]);
#pragma unroll
      for (int c = 0; c < N; c += 4) {
        *reinterpret_cast<float4*>(smB + t * PAD + c) =
            make_float4(v[c], v[c + 1], v[c + 2], v[c + 3]);
      }
      __syncthreads();
      // column-major ownership -> coalesced global stores
#pragma unroll
      for (int r = 0; r < N; ++r) dst[r * N + t] = smB[r * PAD + t];
    }
  }
}

extern "C" void kernel_launch(void* const* d_in, const int* in_sizes, int n_in,
                              void* d_out, int out_size, void* d_ws, size_t ws_size,
                              hipStream_t stream) {
  (void)n_in; (void)out_size; (void)d_ws; (void)ws_size;
  const float* x = (const float*)d_in[0];
  float* out = (float*)d_out;
  const int nmat = in_sizes[0] / (N * N);          // 8192 for the reference shapes
  const int blocks = (nmat + MPB - 1) / MPB;
  sinkhorn_kernel<<<dim3(blocks), dim3(MPB * N), 0, stream>>>(x, out, nmat);
}